// ZeroInitCrossSpeakerEmbeddings_57028575756929
// MI455X (gfx1250) — compile-verified
//
#include <hip/hip_runtime.h>
#include <hip/hip_bf16.h>
#include <math.h>

// ---------------------------------------------------------------------------
// MI455X-targeted implementation.
//
// Roofline: the 65536x1024 fp32 bank (256MB) dominates. One fused streaming
// pass does: row copy -> d_out (the .at[s_i].set clone), row . query logit,
// and online-softmax weighted accumulation (flash-attention style), so the
// bank is read exactly once and written exactly once: ~512MB => ~22us at
// 23.3 TB/s. The 512MB stream exceeds the 192MB L2, so it uses NT temporal
// hints to avoid evicting the MLP weights (37MB, L2-resident across replays).
// All dense GEMVs use V_WMMA_F32_16X16X4_F32 (exact fp32 accumulate).
// ---------------------------------------------------------------------------

typedef __attribute__((ext_vector_type(2))) float v2f;
typedef __attribute__((ext_vector_type(4))) float f4;
typedef __attribute__((ext_vector_type(8))) float v8f;

#define N_SPEAKERS 65536
#define HIDDEN     1024
#define EMBED      1024
#define DIM_UPDATE 2048
#define NPART      1024   // partial blocks from the streaming pass

// ---------------------------------------------------------------------------
// GEMV via V_WMMA_F32_16X16X4_F32.  y[m] = sum_k W[m*K+k] * x[k] (+bias, gelu)
// One wave per 16-row tile. A-matrix ISA layout (16x4 f32):
//   lane l<16 : VGPR0=A[l][0], VGPR1=A[l][1];  lane l+16: A[l][2], A[l][3]
// B (4x16) mirrors the C/D row-pairing: rows (0,2) share VGPR0 across lane
// halves, rows (1,3) share VGPR1. With x replicated across all 16 columns,
// every D column holds identical y-partials; lanes 0 and 16 each own 8 rows.
// ---------------------------------------------------------------------------
__global__ void gemv_wmma_kernel(const float* __restrict__ W,
                                 const float* __restrict__ x,
                                 const float* __restrict__ bias,
                                 float* __restrict__ y,
                                 int K, int apply_gelu) {
  const int lane = threadIdx.x & 31;
  const int wave = threadIdx.x >> 5;
  const int tile = blockIdx.x * (blockDim.x >> 5) + wave;
  const int m0   = tile * 16;
  const int l    = lane & 15;
  const int h    = lane >> 4;            // 0: K 0..1, 1: K 2..3
  const float* wp = W + (size_t)(m0 + l) * (size_t)K + 2 * h;
  const float* xp = x + 2 * h;
  v8f c = {};
#pragma unroll 4
  for (int k0 = 0; k0 < K; k0 += 4) {
    v2f a = *(const v2f*)(wp + k0);
    v2f b = *(const v2f*)(xp + k0);
    c = __builtin_amdgcn_wmma_f32_16x16x4_f32(false, a, false, b,
                                              (short)0, c, false, false);
  }
  if (l == 0) {                          // lanes 0 and 16 hold rows m0.. / m0+8..
    const int mb = m0 + h * 8;
#pragma unroll
    for (int r = 0; r < 8; ++r) {
      float v = c[r];
      if (bias) v += bias[mb + r];
      if (apply_gelu)                    // exact GELU: 0.5*x*(1+erf(x/sqrt(2)))
        v = 0.5f * v * (1.0f + erff(v * 0.70710678118654752f));
      y[mb + r] = v;
    }
  }
}

// ---------------------------------------------------------------------------
// Fused streaming pass: clone bank -> out, online-softmax partials per WG.
// 1024 WGs x 256 thr (8 waves); each wave streams 8 rows with NT b128 loads,
// software-pipelined one row ahead so 8 loads are in flight across the
// cross-lane reduce. Per-WG output: part_m, part_l, part_acc[1024]
// (combined across the 8 waves in LDS).
// ---------------------------------------------------------------------------
__global__ void attn_stream_kernel(const float* __restrict__ bank,
                                   const float* __restrict__ query,
                                   float* __restrict__ out,
                                   float* __restrict__ part_m,
                                   float* __restrict__ part_l,
                                   float* __restrict__ part_acc) {
  __shared__ float s_acc[8 * 1024];   // 32 KB (of 320 KB/WGP)
  __shared__ float s_m[8];
  __shared__ float s_l[8];
  const int tid  = threadIdx.x;
  const int lane = tid & 31;
  const int wave = tid >> 5;

  f4 q[8];
#pragma unroll
  for (int it = 0; it < 8; ++it)
    q[it] = *(const f4*)(query + (it * 32 + lane) * 4);

  float m = -INFINITY, l = 0.0f;
  f4 acc[8];
#pragma unroll
  for (int it = 0; it < 8; ++it) acc[it] = (f4)(0.0f);

  const int rowBase = blockIdx.x * 64 + wave * 8;

  // prologue: NT-load row 0
  f4 v[8];
  {
    const float* src = bank + (size_t)rowBase * 1024;
#pragma unroll
    for (int it = 0; it < 8; ++it)
      v[it] = __builtin_nontemporal_load((const f4*)(src + (it * 32 + lane) * 4));
  }

  for (int r = 0; r < 8; ++r) {
    // issue next row's loads before the dependent reduce (clamped; uniform EXEC)
    const int rn = min(rowBase + r + 1, N_SPEAKERS - 1);
    const float* srcn = bank + (size_t)rn * 1024;
    f4 vn[8];
#pragma unroll
    for (int it = 0; it < 8; ++it)
      vn[it] = __builtin_nontemporal_load((const f4*)(srcn + (it * 32 + lane) * 4));

    // fused clone: NT store of the current row
    float* dst = out + (size_t)(rowBase + r) * 1024;
    float d = 0.0f;
#pragma unroll
    for (int it = 0; it < 8; ++it) {
      __builtin_nontemporal_store(v[it], (f4*)(dst + (it * 32 + lane) * 4));
      d += v[it].x * q[it].x + v[it].y * q[it].y
         + v[it].z * q[it].z + v[it].w * q[it].w;
    }
#pragma unroll
    for (int off = 16; off > 0; off >>= 1)              // wave32 reduce
      d += __shfl_xor(d, off, 32);
    const float mnew  = fmaxf(m, d);
    const float scale = __expf(m - mnew);               // exp(-inf)=0 first row
    const float w     = __expf(d - mnew);
    l = l * scale + w;
#pragma unroll
    for (int it = 0; it < 8; ++it)
      acc[it] = acc[it] * scale + v[it] * w;
    m = mnew;
#pragma unroll
    for (int it = 0; it < 8; ++it) v[it] = vn[it];
  }

  // combine the 8 waves of this WG in LDS
#pragma unroll
  for (int it = 0; it < 8; ++it)
    *(f4*)&s_acc[wave * 1024 + (it * 32 + lane) * 4] = acc[it];
  if (lane == 0) { s_m[wave] = m; s_l[wave] = l; }
  __syncthreads();

  float M = -INFINITY;
#pragma unroll
  for (int w2 = 0; w2 < 8; ++w2) M = fmaxf(M, s_m[w2]);
  float sc[8]; float L = 0.0f;
#pragma unroll
  for (int w2 = 0; w2 < 8; ++w2) {
    sc[w2] = __expf(s_m[w2] - M);
    L += s_l[w2] * sc[w2];
  }
  f4 sum = (f4)(0.0f);
#pragma unroll
  for (int w2 = 0; w2 < 8; ++w2)
    sum += sc[w2] * *(const f4*)&s_acc[w2 * 1024 + tid * 4];
  *(f4*)(part_acc + (size_t)blockIdx.x * 1024 + tid * 4) = sum;
  if (tid == 0) { part_m[blockIdx.x] = M; part_l[blockIdx.x] = L; }
}

// ---------------------------------------------------------------------------
// Combine NPART partial (m,l) into global max/sum; emit per-partial coeff
// coef[p] = exp(m_p - M) / L  so that  context = sum_p coef[p]*acc_p.
// ---------------------------------------------------------------------------
__global__ void reduce_stats_kernel(const float* __restrict__ part_m,
                                    const float* __restrict__ part_l,
                                    float* __restrict__ coef) {
  __shared__ float red[256];
  const int tid = threadIdx.x;
  float m0 = -INFINITY;
  for (int i = tid; i < NPART; i += 256) m0 = fmaxf(m0, part_m[i]);
  red[tid] = m0;
  __syncthreads();
  for (int off = 128; off > 0; off >>= 1) {
    if (tid < off) red[tid] = fmaxf(red[tid], red[tid + off]);
    __syncthreads();
  }
  const float M = red[0];
  __syncthreads();
  float s = 0.0f;
  for (int i = tid; i < NPART; i += 256) s += part_l[i] * __expf(part_m[i] - M);
  red[tid] = s;
  __syncthreads();
  for (int off = 128; off > 0; off >>= 1) {
    if (tid < off) red[tid] += red[tid + off];
    __syncthreads();
  }
  const float invL = 1.0f / red[0];
  for (int i = tid; i < NPART; i += 256)
    coef[i] = __expf(part_m[i] - M) * invL;
}

// ---------------------------------------------------------------------------
// material = concat(u_t, bank[s_i], context) ; context[k] = sum_p coef[p]*acc
// ---------------------------------------------------------------------------
__global__ void build_material_kernel(const float* __restrict__ coef,
                                      const float* __restrict__ part_acc,
                                      const float* __restrict__ u_t,
                                      const float* __restrict__ bank,
                                      const int* __restrict__ s_i_ptr,
                                      float* __restrict__ material) {
  const int k = blockIdx.x * 256 + threadIdx.x;   // 4 blocks => k in [0,1024)
  float s = 0.0f;
#pragma unroll 4
  for (int p = 0; p < NPART; ++p)
    s += coef[p] * part_acc[(size_t)p * 1024 + k];
  material[2048 + k] = s;                          // context
  material[k]        = u_t[k];                     // u_t
  material[1024 + k] = bank[(size_t)(*s_i_ptr) * 1024 + k];  // h_s
}

// ---------------------------------------------------------------------------
// LayerNorm(o) * gamma + beta + h_s  -> overwrite row s_i of the clone.
// ---------------------------------------------------------------------------
__global__ void ln_patch_kernel(const float* __restrict__ bank,
                                const float* __restrict__ obuf,
                                const float* __restrict__ gamma,
                                const float* __restrict__ beta,
                                const int* __restrict__ s_i_ptr,
                                float* __restrict__ out) {
  __shared__ float red[256];
  const int tid = threadIdx.x;
  f4 o = *(const f4*)(obuf + tid * 4);
  red[tid] = o.x + o.y + o.z + o.w;
  __syncthreads();
  for (int off = 128; off > 0; off >>= 1) {
    if (tid < off) red[tid] += red[tid + off];
    __syncthreads();
  }
  const float mu = red[0] * (1.0f / 1024.0f);
  __syncthreads();
  const f4 dv = o - mu;
  red[tid] = dv.x * dv.x + dv.y * dv.y + dv.z * dv.z + dv.w * dv.w;
  __syncthreads();
  for (int off = 128; off > 0; off >>= 1) {
    if (tid < off) red[tid] += red[tid + off];
    __syncthreads();
  }
  const float var = red[0] * (1.0f / 1024.0f);
  const float inv = rsqrtf(var + 1e-5f);
  const size_t base = (size_t)(*s_i_ptr) * 1024 + tid * 4;
  const f4 g  = *(const f4*)(gamma + tid * 4);
  const f4 b  = *(const f4*)(beta  + tid * 4);
  const f4 hs = *(const f4*)(bank + base);
  *(f4*)(out + base) = hs + dv * inv * g + b;
}

// ---------------------------------------------------------------------------
extern "C" void kernel_launch(void* const* d_in, const int* in_sizes, int n_in,
                              void* d_out, int out_size, void* d_ws, size_t ws_size,
                              hipStream_t stream) {
  const float* u_t    = (const float*)d_in[0];
  const float* bank   = (const float*)d_in[1];
  const float* W_attn = (const float*)d_in[2];
  const float* W1     = (const float*)d_in[3];
  const float* b1     = (const float*)d_in[4];
  const float* W2     = (const float*)d_in[5];
  const float* b2     = (const float*)d_in[6];
  const float* gamma  = (const float*)d_in[7];
  const float* beta   = (const float*)d_in[8];
  const int*   s_i    = (const int*)d_in[9];
  float* out = (float*)d_out;
  float* ws  = (float*)d_ws;

  // workspace layout (floats): total ~4.25 MB
  float* query    = ws;            // 1024
  float* part_m   = ws + 1024;     // NPART
  float* part_l   = ws + 2048;     // NPART
  float* coef     = ws + 3072;     // NPART
  float* material = ws + 4096;     // 3072
  float* hbuf     = ws + 7168;     // 2048
  float* obuf     = ws + 9216;     // 1024
  float* part_acc = ws + 10240;    // NPART * 1024

  // 1) query = W_attn @ u_t            (64 tiles, WMMA f32)
  gemv_wmma_kernel<<<16, 128, 0, stream>>>(W_attn, u_t, nullptr, query, EMBED, 0);
  // 2) fused bank clone + online-softmax partials (NT streaming)
  attn_stream_kernel<<<NPART, 256, 0, stream>>>(bank, query, out,
                                                part_m, part_l, part_acc);
  // 3) global softmax stats -> coefficients
  reduce_stats_kernel<<<1, 256, 0, stream>>>(part_m, part_l, coef);
  // 4) context + concat -> material
  build_material_kernel<<<4, 256, 0, stream>>>(coef, part_acc, u_t, bank, s_i,
                                               material);
  // 5) h = gelu(W1 @ material + b1)    (128 tiles, WMMA f32)
  gemv_wmma_kernel<<<32, 128, 0, stream>>>(W1, material, b1, hbuf,
                                           HIDDEN + 2 * EMBED, 1);
  // 6) o = W2 @ h + b2                 (64 tiles, WMMA f32)
  gemv_wmma_kernel<<<16, 128, 0, stream>>>(W2, hbuf, b2, obuf, DIM_UPDATE, 0);
  // 7) LayerNorm + residual, patch row s_i (after the clone, stream-ordered)
  ln_patch_kernel<<<1, 256, 0, stream>>>(bank, obuf, gamma, beta, s_i, out);
}